// HypergraphConv_42442866819268
// MI455X (gfx1250) — compile-verified
//
#include <hip/hip_runtime.h>
#include <math.h>

// ---------------------------------------------------------------------------
// HypergraphConv sparse-attention forward for MI455X (gfx1250, wave32).
//   h   = x @ W + b                      [8192 x 256]   (WMMA f32 16x16x4)
//   p   = hflat @ cat(W1a|W1b)           [16384 x 128]  (WMMA f32 16x16x4)
//   sc  = leakyrelu(p_src[s]+p_dst[d])·w2 per edge/head
//   att = softmax over ALL edges per head (max + exp + sum, tree reductions)
//   out[n] = mean_h( sum_{e: src=n} att[e,h] * h[dst[e], h, :] )
// Edges arrive sorted by src -> per-node aggregation via binary search,
// no atomics anywhere => deterministic, graph-capture safe.
// Each GEMM wave computes a 16(M) x 64(N) strip: one A fragment feeds 4 WMMAs
// per K-step (4x A reuse, 4x WMMA density vs. a single 16x16 tile).
// ---------------------------------------------------------------------------

#define NN 8192
#define KDIM 128          // in_channels == per-head dim
#define HCOLS 256         // OUT*HEADS
#define PROWS (NN * 2)    // 16384 flattened (node, head) rows
#define PCOLS 128         // 64 src-proj cols | 64 dst-proj cols
#define NB 256            // reduction blocks
#define LRELU 0.01f

typedef __attribute__((ext_vector_type(2))) float v2f;
typedef __attribute__((ext_vector_type(8))) float v8f;

// --------------------------- GEMM 1: h = x@W + b ---------------------------
__global__ void gemm_h_kernel(const float* __restrict__ x,     // [8192,128]
                              const float* __restrict__ w,     // [128,256]
                              const float* __restrict__ bias,  // [256]
                              float* __restrict__ h) {         // [8192,256]
    const int lane = threadIdx.x & 31;
    const int wave = threadIdx.x >> 5;
    const int strip = blockIdx.x * (blockDim.x >> 5) + wave;   // 0..2047
    const int m0 = (strip >> 2) * 16;      // 512 strips along M
    const int n0 = (strip & 3) * 64;       // 4 strips of 64 along N (256 total)
    const int lh = lane & 15;              // lane within half-wave
    const int koff = (lane >> 4) * 2;      // lanes 0-15 -> K=0,1 ; 16-31 -> K=2,3

    v8f acc0 = {}, acc1 = {}, acc2 = {}, acc3 = {};
#pragma unroll 4
    for (int k = 0; k < KDIM; k += 4) {
        // A fragment: 16x4 tile of x (row m0+lh, two consecutive K values)
        const float* ap = x + (m0 + lh) * KDIM + k + koff;
        v2f a; a.x = ap[0]; a.y = ap[1];
        // B fragments: 4x16 tiles of W at n0, n0+16, n0+32, n0+48
        const float* b0p = w + (k + koff) * HCOLS + n0 + lh;
        const float* b1p = w + (k + koff + 1) * HCOLS + n0 + lh;
        v2f b;
        b.x = b0p[0];  b.y = b1p[0];
        acc0 = __builtin_amdgcn_wmma_f32_16x16x4_f32(false, a, false, b, (short)0, acc0, false, false);
        b.x = b0p[16]; b.y = b1p[16];
        acc1 = __builtin_amdgcn_wmma_f32_16x16x4_f32(false, a, false, b, (short)0, acc1, false, false);
        b.x = b0p[32]; b.y = b1p[32];
        acc2 = __builtin_amdgcn_wmma_f32_16x16x4_f32(false, a, false, b, (short)0, acc2, false, false);
        b.x = b0p[48]; b.y = b1p[48];
        acc3 = __builtin_amdgcn_wmma_f32_16x16x4_f32(false, a, false, b, (short)0, acc3, false, false);
    }
    // C/D layout: VGPR j -> row m0+j (lanes 0-15) / m0+8+j (lanes 16-31)
    const int rbase = m0 + (lane >> 4) * 8;
#pragma unroll
    for (int j = 0; j < 8; ++j) {
        float* hp = h + (rbase + j) * HCOLS + n0 + lh;
        hp[0]  = acc0[j] + bias[n0 + lh];
        hp[16] = acc1[j] + bias[n0 + 16 + lh];
        hp[32] = acc2[j] + bias[n0 + 32 + lh];
        hp[48] = acc3[j] + bias[n0 + 48 + lh];
    }
}

// ---------------- GEMM 2: p = hflat @ cat(W1a | W1b) -----------------------
// hflat is h reinterpreted as [16384,128]; B column n<64 comes from
// att_weight1 rows 0..127, n>=64 from rows 128..255.
__device__ __forceinline__ float w1cat(const float* __restrict__ w1, int k, int n) {
    return (n < 64) ? w1[k * 64 + n] : w1[(128 + k) * 64 + (n - 64)];
}

__global__ void gemm_p_kernel(const float* __restrict__ hflat,  // [16384,128]
                              const float* __restrict__ w1,     // [256,64]
                              float* __restrict__ p) {          // [16384,128]
    const int lane = threadIdx.x & 31;
    const int wave = threadIdx.x >> 5;
    const int strip = blockIdx.x * (blockDim.x >> 5) + wave;    // 0..2047
    const int m0 = (strip >> 1) * 16;       // 1024 strips along M
    const int n0 = (strip & 1) * 64;        // 2 strips of 64 along N (128 total)
    const int lh = lane & 15;
    const int koff = (lane >> 4) * 2;

    v8f acc0 = {}, acc1 = {}, acc2 = {}, acc3 = {};
#pragma unroll 4
    for (int k = 0; k < KDIM; k += 4) {
        const float* ap = hflat + (m0 + lh) * KDIM + k + koff;
        v2f a; a.x = ap[0]; a.y = ap[1];
        v2f b;
        b.x = w1cat(w1, k + koff, n0 + lh);
        b.y = w1cat(w1, k + koff + 1, n0 + lh);
        acc0 = __builtin_amdgcn_wmma_f32_16x16x4_f32(false, a, false, b, (short)0, acc0, false, false);
        b.x = w1cat(w1, k + koff, n0 + 16 + lh);
        b.y = w1cat(w1, k + koff + 1, n0 + 16 + lh);
        acc1 = __builtin_amdgcn_wmma_f32_16x16x4_f32(false, a, false, b, (short)0, acc1, false, false);
        b.x = w1cat(w1, k + koff, n0 + 32 + lh);
        b.y = w1cat(w1, k + koff + 1, n0 + 32 + lh);
        acc2 = __builtin_amdgcn_wmma_f32_16x16x4_f32(false, a, false, b, (short)0, acc2, false, false);
        b.x = w1cat(w1, k + koff, n0 + 48 + lh);
        b.y = w1cat(w1, k + koff + 1, n0 + 48 + lh);
        acc3 = __builtin_amdgcn_wmma_f32_16x16x4_f32(false, a, false, b, (short)0, acc3, false, false);
    }
    const int rbase = m0 + (lane >> 4) * 8;
#pragma unroll
    for (int j = 0; j < 8; ++j) {
        float* pp = p + (rbase + j) * PCOLS + n0 + lh;
        pp[0]  = acc0[j];
        pp[16] = acc1[j];
        pp[32] = acc2[j];
        pp[48] = acc3[j];
    }
}

// ------------------------- Per-edge attention scores -----------------------
__global__ void score_kernel(const float* __restrict__ p,    // [16384,128]
                             const int* __restrict__ src,
                             const int* __restrict__ dst,
                             const float* __restrict__ w2,   // [64]
                             float* __restrict__ scores,     // [E*2]
                             int E) {
    int idx = blockIdx.x * blockDim.x + threadIdx.x;
    if (idx >= E * 2) return;
    int e = idx >> 1, hd = idx & 1;
    int s = src[e], d = dst[e];
    const float4* ps = reinterpret_cast<const float4*>(p + (s * 2 + hd) * PCOLS);      // cols 0..63
    const float4* pd = reinterpret_cast<const float4*>(p + (d * 2 + hd) * PCOLS + 64); // cols 64..127
    const float4* wv = reinterpret_cast<const float4*>(w2);
    float sc = 0.f;
#pragma unroll 4
    for (int f = 0; f < 16; ++f) {
        float4 a = ps[f], b = pd[f], c = wv[f];
        float v;
        v = a.x + b.x; v = (v > 0.f) ? v : LRELU * v; sc = fmaf(v, c.x, sc);
        v = a.y + b.y; v = (v > 0.f) ? v : LRELU * v; sc = fmaf(v, c.y, sc);
        v = a.z + b.z; v = (v > 0.f) ? v : LRELU * v; sc = fmaf(v, c.z, sc);
        v = a.w + b.w; v = (v > 0.f) ? v : LRELU * v; sc = fmaf(v, c.w, sc);
    }
    scores[idx] = sc;
}

// ------------------- Deterministic global softmax reductions ---------------
__global__ void max_partial_kernel(const float* __restrict__ scores, int E,
                                   float* __restrict__ partials) {  // [NB*2]
    __shared__ float sm[256];
    for (int hd = 0; hd < 2; ++hd) {
        float m = -3.0e38f;
        for (int e = blockIdx.x * blockDim.x + threadIdx.x; e < E;
             e += gridDim.x * blockDim.x)
            m = fmaxf(m, scores[e * 2 + hd]);
        sm[threadIdx.x] = m;
        __syncthreads();
        for (int s = 128; s > 0; s >>= 1) {
            if (threadIdx.x < s)
                sm[threadIdx.x] = fmaxf(sm[threadIdx.x], sm[threadIdx.x + s]);
            __syncthreads();
        }
        if (threadIdx.x == 0) partials[blockIdx.x * 2 + hd] = sm[0];
        __syncthreads();
    }
}

__global__ void max_final_kernel(const float* __restrict__ partials,  // [NB*2]
                                 float* __restrict__ gmax) {          // [2]
    __shared__ float sm[256];
    for (int hd = 0; hd < 2; ++hd) {
        sm[threadIdx.x] = partials[threadIdx.x * 2 + hd];
        __syncthreads();
        for (int s = 128; s > 0; s >>= 1) {
            if (threadIdx.x < s)
                sm[threadIdx.x] = fmaxf(sm[threadIdx.x], sm[threadIdx.x + s]);
            __syncthreads();
        }
        if (threadIdx.x == 0) gmax[hd] = sm[0];
        __syncthreads();
    }
}

__global__ void exp_sum_partial_kernel(float* __restrict__ scores, int E,
                                       const float* __restrict__ gmax,
                                       float* __restrict__ partials) {  // [NB*2]
    __shared__ float sm[256];
    for (int hd = 0; hd < 2; ++hd) {
        const float mx = gmax[hd];
        float acc = 0.f;
        for (int e = blockIdx.x * blockDim.x + threadIdx.x; e < E;
             e += gridDim.x * blockDim.x) {
            float v = expf(scores[e * 2 + hd] - mx);
            scores[e * 2 + hd] = v;   // in-place -> exp values
            acc += v;
        }
        sm[threadIdx.x] = acc;
        __syncthreads();
        for (int s = 128; s > 0; s >>= 1) {
            if (threadIdx.x < s) sm[threadIdx.x] += sm[threadIdx.x + s];
            __syncthreads();
        }
        if (threadIdx.x == 0) partials[blockIdx.x * 2 + hd] = sm[0];
        __syncthreads();
    }
}

__global__ void sum_final_kernel(const float* __restrict__ partials,
                                 float* __restrict__ gsum) {  // [2]
    __shared__ float sm[256];
    for (int hd = 0; hd < 2; ++hd) {
        sm[threadIdx.x] = partials[threadIdx.x * 2 + hd];
        __syncthreads();
        for (int s = 128; s > 0; s >>= 1) {
            if (threadIdx.x < s) sm[threadIdx.x] += sm[threadIdx.x + s];
            __syncthreads();
        }
        if (threadIdx.x == 0) gsum[hd] = sm[0];
        __syncthreads();
    }
}

// ------------------ Per-src-node aggregation (edges sorted) ----------------
__global__ void agg_kernel(const float* __restrict__ h,      // [8192,256]
                           const float* __restrict__ expv,   // [E*2]
                           const float* __restrict__ gsum,   // [2]
                           const int* __restrict__ src,
                           const int* __restrict__ dst,
                           int E,
                           float* __restrict__ out) {        // [8192,128]
    const int n = blockIdx.x;
    __shared__ int rng[2];
    if (threadIdx.x == 0) {
        int lo = 0, hi = E;                     // lower_bound(src, n)
        while (lo < hi) { int m = (lo + hi) >> 1; if (src[m] < n) lo = m + 1; else hi = m; }
        rng[0] = lo;
        int lo2 = lo; hi = E;                   // lower_bound(src, n+1)
        while (lo2 < hi) { int m = (lo2 + hi) >> 1; if (src[m] < n + 1) lo2 = m + 1; else hi = m; }
        rng[1] = lo2;
    }
    __syncthreads();
    const int c = threadIdx.x;                  // 0..127
    const float inv0 = 1.f / gsum[0], inv1 = 1.f / gsum[1];
    const float2* ev = reinterpret_cast<const float2*>(expv);
    float a0 = 0.f, a1 = 0.f;
    for (int e = rng[0]; e < rng[1]; ++e) {
        const int d = dst[e];
        const float2 wgt = ev[e];               // {exp_h0, exp_h1}
        a0 = fmaf(wgt.x, h[d * HCOLS + c], a0);
        a1 = fmaf(wgt.y, h[d * HCOLS + 128 + c], a1);
    }
    out[n * KDIM + c] = 0.5f * (a0 * inv0 + a1 * inv1);
}

// ---------------------------------------------------------------------------
extern "C" void kernel_launch(void* const* d_in, const int* in_sizes, int n_in,
                              void* d_out, int out_size, void* d_ws, size_t ws_size,
                              hipStream_t stream) {
    const float* x    = (const float*)d_in[0];
    // d_in[1] = theta (unused; edge_index already encodes sparsity)
    const float* w    = (const float*)d_in[2];
    const float* bias = (const float*)d_in[3];
    const float* w1   = (const float*)d_in[4];
    const float* w2   = (const float*)d_in[5];
    const int*   ei   = (const int*)d_in[6];
    const int E = in_sizes[6] / 2;
    const int* src = ei;         // edge_index[0]  (sorted ascending)
    const int* dst = ei + E;     // edge_index[1]
    float* out = (float*)d_out;

    // Workspace carve-up (floats)
    float* ws = (float*)d_ws;
    float* h        = ws;                               // 8192*256
    float* p        = h + NN * HCOLS;                   // 16384*128
    float* scores   = p + PROWS * PCOLS;                // E*2 (scores -> exp)
    float* partials = scores + (size_t)E * 2;           // NB*2
    float* gmax     = partials + NB * 2;                // 2
    float* gsum     = gmax + 2;                         // 2

    // 1) h = x@W + b : 512 M-strips x 4 N-strips = 2048 waves, 4 waves/block
    gemm_h_kernel<<<512, 128, 0, stream>>>(x, w, bias, h);
    // 2) p = hflat @ cat(W1a|W1b) : 1024 M-strips x 2 N-strips = 2048 waves
    gemm_p_kernel<<<512, 128, 0, stream>>>(h, w1, p);
    // 3) per-edge scores
    {
        int thr = E * 2;
        score_kernel<<<(thr + 255) / 256, 256, 0, stream>>>(p, src, dst, w2, scores, E);
    }
    // 4-5) global max per head
    max_partial_kernel<<<NB, 256, 0, stream>>>(scores, E, partials);
    max_final_kernel<<<1, 256, 0, stream>>>(partials, gmax);
    // 6-7) exp in-place + global sum per head
    exp_sum_partial_kernel<<<NB, 256, 0, stream>>>(scores, E, gmax, partials);
    sum_final_kernel<<<1, 256, 0, stream>>>(partials, gsum);
    // 8) aggregate per src node, mean over heads
    agg_kernel<<<NN, 128, 0, stream>>>(h, scores, gsum, src, dst, E, out);
}